// GNN_60069412602437
// MI455X (gfx1250) — compile-verified
//
#include <hip/hip_runtime.h>
#include <hip/hip_bf16.h>

#define EMB   300
#define LDH   304   // h / accumulator leading dim (K padded to 304 = 76*4)
#define LDC   320   // hW leading dim (N padded to 320 = 5 groups of 64)
#define EPSV  1e-5f

typedef float v2f __attribute__((ext_vector_type(2)));
typedef float v8f __attribute__((ext_vector_type(8)));

// ---------------- h0 = emb[atom] + x_feat @ x_weight  (padded to LDH, pad cols = 0)
__global__ void gnn_h_init(const int* __restrict__ atom, const float* __restrict__ xf,
                           const float* __restrict__ emb, const float* __restrict__ xw,
                           float* __restrict__ h, int N, int PAD, int MP) {
    size_t idx = (size_t)blockIdx.x * blockDim.x + threadIdx.x;
    size_t tot = (size_t)MP * LDH;
    if (idx >= tot) return;
    int n = (int)(idx / LDH), j = (int)(idx % LDH);
    float v = 0.0f;
    if (n < N && j < EMB) {
        v = emb[(size_t)atom[n] * EMB + j];
        const float* xr = xf + (size_t)n * PAD;
        for (int k = 0; k < PAD; ++k) v = fmaf(xr[k], xw[k * EMB + j], v);
    }
    h[idx] = v;
}

// ---------------- pack mlp_w[l] (300x300) into zero-padded [304][320]
__global__ void gnn_pack_w(const float* __restrict__ W, float* __restrict__ Wp) {
    int idx = blockIdx.x * blockDim.x + threadIdx.x;
    if (idx >= LDH * LDC) return;
    int k = idx / LDC, j = idx % LDC;
    Wp[idx] = (k < EMB && j < EMB) ? W[k * EMB + j] : 0.0f;
}

// ---------------- p0 = w_e0 @ W, p1 = w_e1 @ W, selfvec = 4*p1 + b
__global__ void gnn_pvec(const float* __restrict__ ew, const float* __restrict__ W,
                         const float* __restrict__ b,
                         float* __restrict__ p0, float* __restrict__ p1,
                         float* __restrict__ sv) {
    int j = threadIdx.x;
    if (j >= EMB) return;
    float s0 = 0.0f, s1 = 0.0f;
    for (int k = 0; k < EMB; ++k) {
        float w = W[k * EMB + j];
        s0 = fmaf(ew[k],       w, s0);
        s1 = fmaf(ew[EMB + k], w, s1);
    }
    p0[j] = s0;
    p1[j] = s1;
    sv[j] = fmaf(4.0f, s1, b[j]);
}

// ---------------- hW = h @ Wp via V_WMMA_F32_16X16X4_F32; wave computes 16x64 tile
__global__ void __launch_bounds__(256)
gnn_gemm_wmma(const float* __restrict__ A, const float* __restrict__ W,
              float* __restrict__ C, int mtiles) {
    int wave = (int)((blockIdx.x * blockDim.x + threadIdx.x) >> 5);
    int lane = threadIdx.x & 31;
    int tasks = mtiles * 5;                 // 5 column groups of 64 (N padded to 320)
    if (wave >= tasks) return;              // whole-wave uniform exit: EXEC all 1s for WMMA
    int mt = wave / 5, ng = wave % 5;
    int m0 = mt << 4, n0 = ng << 6;
    int laneLow = lane & 15;
    int hi = lane >> 4;                     // lanes 16-31 hold K+2,K+3 (A) / rows K+2,K+3 (B)

    v8f acc0 = {}, acc1 = {}, acc2 = {}, acc3 = {};
    const float* arow = A + (size_t)(m0 + laneLow) * LDH + (hi << 1);

    for (int kk = 0; kk < LDH; kk += 4) {
        v2f a = *(const v2f*)(arow + kk);   // A[m0+laneLow][kk+hi*2 .. +1], 8B aligned
        int kr = kk + (hi << 1);
        const float* w0 = W + (size_t)kr * LDC + n0 + laneLow;
        const float* w1 = w0 + LDC;
        v2f b0 = { w0[0],  w1[0]  };
        v2f b1 = { w0[16], w1[16] };
        v2f b2 = { w0[32], w1[32] };
        v2f b3 = { w0[48], w1[48] };
        acc0 = __builtin_amdgcn_wmma_f32_16x16x4_f32(false, a, false, b0, (short)0, acc0, false, false);
        acc1 = __builtin_amdgcn_wmma_f32_16x16x4_f32(false, a, false, b1, (short)0, acc1, false, false);
        acc2 = __builtin_amdgcn_wmma_f32_16x16x4_f32(false, a, false, b2, (short)0, acc2, false, false);
        acc3 = __builtin_amdgcn_wmma_f32_16x16x4_f32(false, a, false, b3, (short)0, acc3, false, false);
    }
    // C/D layout: VGPR v -> M = v (lanes 0-15) / v+8 (lanes 16-31), N = laneLow
    float* crow = C + (size_t)(m0 + (hi << 3)) * LDC + n0 + laneLow;
#pragma unroll
    for (int v = 0; v < 8; ++v) {
        crow[(size_t)v * LDC + 0]  = acc0[v];
        crow[(size_t)v * LDC + 16] = acc1[v];
        crow[(size_t)v * LDC + 32] = acc2[v];
        crow[(size_t)v * LDC + 48] = acc3[v];
    }
}

// ---------------- zero a float buffer
__global__ void gnn_zero(float* __restrict__ p, size_t n) {
    size_t idx = (size_t)blockIdx.x * blockDim.x + threadIdx.x;
    if (idx < n) p[idx] = 0.0f;
}

// ---------------- per-edge: relu(hW[row] + ea0*p0 + ea1*p1 + b) scatter-added at col
__global__ void gnn_edge_scatter(const int* __restrict__ rowi, const int* __restrict__ coli,
                                 const float* __restrict__ ea, const float* __restrict__ hW,
                                 const float* __restrict__ p0, const float* __restrict__ p1,
                                 const float* __restrict__ bias, float* __restrict__ hacc,
                                 int E) {
    int wave = (int)(((size_t)blockIdx.x * blockDim.x + threadIdx.x) >> 5);
    int lane = threadIdx.x & 31;
    if (wave >= E) return;
    int r = rowi[wave], c = coli[wave];
    float a0 = ea[2 * (size_t)wave], a1 = ea[2 * (size_t)wave + 1];
    const float* src = hW + (size_t)r * LDC;
    float* dst = hacc + (size_t)c * LDH;
    for (int j = lane; j < EMB; j += 32) {
        float m = fmaf(a0, p0[j], fmaf(a1, p1[j], src[j] + bias[j]));
        m = fmaxf(m, 0.0f);
        atomicAdd(dst + j, m);
    }
}

// ---------------- self-loop: h[n] += relu(hW[n] + selfvec) (runs after edge kernel)
__global__ void gnn_self_add(const float* __restrict__ hW, const float* __restrict__ sv,
                             float* __restrict__ h, int N) {
    size_t idx = (size_t)blockIdx.x * blockDim.x + threadIdx.x;
    if (idx >= (size_t)N * EMB) return;
    int n = (int)(idx / EMB), j = (int)(idx % EMB);
    float m = fmaxf(hW[(size_t)n * LDC + j] + sv[j], 0.0f);
    h[(size_t)n * LDH + j] += m;
}

// ---------------- batchnorm: column sum / sumsq reduction
__global__ void gnn_bn_reduce(const float* __restrict__ h, float* __restrict__ csum,
                              float* __restrict__ csq, int N) {
    int t = threadIdx.x;
    if (t >= EMB) return;
    int rows = (N + gridDim.x - 1) / gridDim.x;
    int r0 = blockIdx.x * rows;
    int r1 = r0 + rows; if (r1 > N) r1 = N;
    float s = 0.0f, q = 0.0f;
    for (int r = r0; r < r1; ++r) {
        float v = h[(size_t)r * LDH + t];
        s += v;
        q = fmaf(v, v, q);
    }
    atomicAdd(csum + t, s);
    atomicAdd(csq + t, q);
}

__global__ void gnn_bn_coeff(const float* __restrict__ csum, const float* __restrict__ csq,
                             const float* __restrict__ gamma, const float* __restrict__ beta,
                             float* __restrict__ scl, float* __restrict__ shf, int N) {
    int j = threadIdx.x;
    if (j >= EMB) return;
    float invN = 1.0f / (float)N;
    float mu  = csum[j] * invN;
    float var = csq[j] * invN - mu * mu;      // biased variance
    float s = rsqrtf(var + EPSV) * gamma[j];
    scl[j] = s;
    shf[j] = fmaf(-mu, s, beta[j]);
}

__global__ void gnn_bn_apply(float* __restrict__ h, const float* __restrict__ scl,
                             const float* __restrict__ shf, int N, int do_relu) {
    size_t idx = (size_t)blockIdx.x * blockDim.x + threadIdx.x;
    if (idx >= (size_t)N * EMB) return;
    int n = (int)(idx / EMB), j = (int)(idx % EMB);
    float v = fmaf(h[(size_t)n * LDH + j], scl[j], shf[j]);
    if (do_relu) v = fmaxf(v, 0.0f);
    h[(size_t)n * LDH + j] = v;
}

// ---------------- strip padding into d_out
__global__ void gnn_copy_out(const float* __restrict__ h, float* __restrict__ out, int N) {
    size_t idx = (size_t)blockIdx.x * blockDim.x + threadIdx.x;
    if (idx >= (size_t)N * EMB) return;
    out[idx] = h[(idx / EMB) * LDH + (idx % EMB)];
}

extern "C" void kernel_launch(void* const* d_in, const int* in_sizes, int n_in,
                              void* d_out, int out_size, void* d_ws, size_t ws_size,
                              hipStream_t stream) {
    const int*   atom  = (const int*)d_in[0];
    const float* xf    = (const float*)d_in[1];
    const int*   eidx  = (const int*)d_in[2];
    const float* eattr = (const float*)d_in[3];
    // d_in[4] (batch) and d_in[5] (metal_feature) feed dead code in the reference
    const float* emb   = (const float*)d_in[6];
    const float* xw    = (const float*)d_in[7];
    const float* ew    = (const float*)d_in[8];
    const float* mw    = (const float*)d_in[9];
    const float* mb    = (const float*)d_in[10];
    const float* gamma = (const float*)d_in[11];
    const float* beta  = (const float*)d_in[12];

    const int N   = in_sizes[0];
    const int E   = in_sizes[2] / 2;
    const int PAD = in_sizes[1] / N;
    const int L   = in_sizes[10] / EMB;
    const int MP  = (N + 15) & ~15;     // 50000 is already a multiple of 16

    float* h    = (float*)d_ws;                 // [MP][LDH]
    float* hW   = h  + (size_t)MP * LDH;        // [MP][LDC]
    float* Wp   = hW + (size_t)MP * LDC;        // [LDH][LDC]
    float* p0   = Wp + (size_t)LDH * LDC;       // 320
    float* p1   = p0 + LDC;
    float* sv   = p1 + LDC;
    float* csum = sv + LDC;                     // csum(320) + csq(320) contiguous
    float* csq  = csum + LDC;
    float* scl  = csq + LDC;
    float* shf  = scl + LDC;

    const int* rowi = eidx;
    const int* coli = eidx + E;

    const int T = 256;
    const size_t hElems  = (size_t)MP * LDH;
    const size_t neElems = (size_t)N * EMB;

    gnn_h_init<<<(unsigned)((hElems + T - 1) / T), T, 0, stream>>>(atom, xf, emb, xw, h, N, PAD, MP);

    const int mtiles = MP / 16;
    const unsigned ggrid = (unsigned)((mtiles * 5 + 7) / 8);   // 8 waves per block

    for (int l = 0; l < L; ++l) {
        const float* Wl = mw + (size_t)l * EMB * EMB;
        gnn_pack_w<<<(LDH * LDC + T - 1) / T, T, 0, stream>>>(Wl, Wp);
        gnn_pvec<<<1, 320, 0, stream>>>(ew + (size_t)l * 2 * EMB, Wl, mb + (size_t)l * EMB,
                                        p0, p1, sv);
        gnn_gemm_wmma<<<ggrid, T, 0, stream>>>(h, Wp, hW, mtiles);
        gnn_zero<<<(unsigned)((hElems + T - 1) / T), T, 0, stream>>>(h, hElems);
        gnn_edge_scatter<<<(unsigned)(((size_t)E * 32 + T - 1) / T), T, 0, stream>>>(
            rowi, coli, eattr, hW, p0, p1, mb + (size_t)l * EMB, h, E);
        gnn_self_add<<<(unsigned)((neElems + T - 1) / T), T, 0, stream>>>(hW, sv, h, N);
        gnn_zero<<<3, T, 0, stream>>>(csum, 2 * LDC);
        gnn_bn_reduce<<<200, 320, 0, stream>>>(h, csum, csq, N);
        gnn_bn_coeff<<<1, 320, 0, stream>>>(csum, csq, gamma + (size_t)l * EMB,
                                            beta + (size_t)l * EMB, scl, shf, N);
        gnn_bn_apply<<<(unsigned)((neElems + T - 1) / T), T, 0, stream>>>(
            h, scl, shf, N, (l < L - 1) ? 1 : 0);
    }

    gnn_copy_out<<<(unsigned)((neElems + T - 1) / T), T, 0, stream>>>(h, (float*)d_out, N);
}